// GCN_75720273428871
// MI455X (gfx1250) — compile-verified
//
#include <hip/hip_runtime.h>
#include <math.h>

#define N_NODES 100000
#define N_EDGES 1600000
#define NFEAT   128
#define NHID    64
#define NCLASS  40
#define NC_PAD  48      // NCLASS padded to a multiple of 16 for the layer-3 GEMM

typedef float v2f __attribute__((ext_vector_type(2)));
typedef float v8f __attribute__((ext_vector_type(8)));

// ---------------------------------------------------------------- utilities
__global__ void k_zero_i32(int* __restrict__ p, int n) {
    int i = blockIdx.x * blockDim.x + threadIdx.x;
    if (i < n) p[i] = 0;
}

__global__ void k_hist(const int* __restrict__ dst, int* __restrict__ cnt, int e) {
    int i = blockIdx.x * blockDim.x + threadIdx.x;
    if (i < e) atomicAdd(&cnt[dst[i]], 1);
}

// exclusive scan, 1024 elements per block (Hillis-Steele in LDS)
__global__ void k_scan_block(const int* __restrict__ cnt, int* __restrict__ rowptr,
                             int* __restrict__ bsum, int n) {
    __shared__ int sh[1024];
    int tid = threadIdx.x;
    int i   = blockIdx.x * 1024 + tid;
    int v   = (i < n) ? cnt[i] : 0;
    sh[tid] = v;
    __syncthreads();
    for (int off = 1; off < 1024; off <<= 1) {
        int t = (tid >= off) ? sh[tid - off] : 0;
        __syncthreads();
        sh[tid] += t;
        __syncthreads();
    }
    if (i < n) rowptr[i] = sh[tid] - v;           // exclusive
    if (tid == 1023) bsum[blockIdx.x] = sh[1023]; // block total
}

__global__ void k_scan_sums(int* __restrict__ bsum, int nb) {
    if (threadIdx.x == 0 && blockIdx.x == 0) {
        int acc = 0;
        for (int i = 0; i < nb; ++i) { int t = bsum[i]; bsum[i] = acc; acc += t; }
    }
}

__global__ void k_scan_add(int* __restrict__ rowptr, const int* __restrict__ bsum,
                           int n, int total) {
    int i = blockIdx.x * blockDim.x + threadIdx.x;
    if (i < n)       rowptr[i] += bsum[i >> 10];
    else if (i == n) rowptr[n]  = total;
}

__global__ void k_prep(const int* __restrict__ cnt, const int* __restrict__ rowptr,
                       float* __restrict__ dinv, int* __restrict__ cursor, int n) {
    int i = blockIdx.x * blockDim.x + threadIdx.x;
    if (i < n) {
        dinv[i]   = rsqrtf((float)cnt[i] + 1.0f);  // +1 self-loop; deg>=1 always
        cursor[i] = rowptr[i];
    }
}

// CSR fill: pack (src, norm-weight) into a single int2 per edge (one b64 load later)
__global__ void k_fill(const int* __restrict__ src, const int* __restrict__ dst,
                       const float* __restrict__ dinv, int* __restrict__ cursor,
                       int2* __restrict__ cedge, int e) {
    int i = blockIdx.x * blockDim.x + threadIdx.x;
    if (i < e) {
        int s = src[i], d = dst[i];
        int p = atomicAdd(&cursor[d], 1);
        int2 pk;
        pk.x = s;
        pk.y = __float_as_int(dinv[s] * dinv[d]);
        cedge[p] = pk;
    }
}

// Pack W[K,Nin] row-major into B-fragment order, zero-padding columns to Nout:
// Wp viewed as float2[(K/2) * Nout]; Wp2[k2*Nout + n] = { W[2k2][n], W[2k2+1][n] }
__global__ void k_packW(const float* __restrict__ Win, float* __restrict__ Wp,
                        int K, int Nin, int Nout) {
    int i = blockIdx.x * blockDim.x + threadIdx.x;
    if (i >= K * Nout) return;
    int k = i / Nout, n = i - k * Nout;
    float v = (n < Nin) ? Win[(size_t)k * Nin + n] : 0.f;
    Wp[(((size_t)(k >> 1) * Nout + n) << 1) + (k & 1)] = v;
}

// ---------------------------------------------------------------- WMMA GEMM
// H[M,NOUT] = act(A[M,K]) @ W[K,NOUT]  via V_WMMA_F32_16X16X4_F32 (fp32-exact).
// One wave: 5 independent 16x16 M-subtiles x one 16-col N-tile.
//   - one b64 B-frag load per k-step, reused by 5 WMMAs
//   - 5 independent accumulator chains (no WMMA->WMMA RAW stalls)
//   - all control flow wave-uniform; EXEC stays all-ones (WMMA requirement)
// A-frag (16x4): lanes 0-15 -> K={k,k+1}; lanes 16-31 -> K={k+2,k+3} (regs 0,1).
// C/D: reg r -> row m0 + half*8 + r, col n0 + l.
template <int K, int NOUT, bool RELU>
__global__ void k_gemm_wmma(const float* __restrict__ A, const float* __restrict__ Wp,
                            float* __restrict__ H) {
    constexpr int NT = NOUT / 16;
    constexpr int MT = 5;                           // 6250 row-tiles / 5 = 1250 exactly
    constexpr int MGROUPS = N_NODES / (16 * MT);
    int wave = (blockIdx.x * blockDim.x + threadIdx.x) >> 5;
    int lane = threadIdx.x & 31;
    int group = wave / NT;
    int nTile = wave - group * NT;
    if (group >= MGROUPS) return;                   // wave-uniform
    const int m0 = group * (16 * MT);
    const int n0 = nTile * 16;
    const int half = lane >> 4, l = lane & 15;
    const int colB = n0 + l;

    const float2* wp = (const float2*)Wp;
    const float* arow[MT];
#pragma unroll
    for (int t = 0; t < MT; ++t) arow[t] = A + (size_t)(m0 + t * 16 + l) * K;

    v8f c[MT];
    const v8f zero = {0.f, 0.f, 0.f, 0.f, 0.f, 0.f, 0.f, 0.f};
#pragma unroll
    for (int t = 0; t < MT; ++t) c[t] = zero;

#pragma unroll 4
    for (int k = 0; k < K; k += 4) {
        float2 bb = wp[(size_t)((k >> 1) + half) * NOUT + colB];
        v2f b; b.x = bb.x; b.y = bb.y;
#pragma unroll
        for (int t = 0; t < MT; ++t) {
            float2 av = *(const float2*)(arow[t] + k + 2 * half);
            if (RELU) { av.x = fmaxf(av.x, 0.f); av.y = fmaxf(av.y, 0.f); }
            v2f a; a.x = av.x; a.y = av.y;
            c[t] = __builtin_amdgcn_wmma_f32_16x16x4_f32(
                       false, a, false, b, (short)0, c[t], false, false);
        }
    }
#pragma unroll
    for (int t = 0; t < MT; ++t) {
#pragma unroll
        for (int r = 0; r < 8; ++r) {
            int row = m0 + t * 16 + half * 8 + r;
            H[(size_t)row * NOUT + colB] = c[t][r];
        }
    }
}

// ------------------------------------------------------- CSR gather (no atomics)
// out[i] = b + h[i]*dinv[i]^2 + sum_{e in CSR(i)} h[src[e]] * w[e]
__global__ void k_agg64(const float* __restrict__ H, const int* __restrict__ rowptr,
                        const int2* __restrict__ cedge, const float* __restrict__ dinv,
                        const float* __restrict__ bias, float* __restrict__ OUT, int n) {
    int node = (blockIdx.x * blockDim.x + threadIdx.x) >> 5;
    int lane = threadIdx.x & 31;
    if (node >= n) return;
    node = __builtin_amdgcn_readfirstlane(node);   // provably uniform -> scalar loads
    float di = dinv[node];
    float wself = di * di;
    const float2* hp = (const float2*)H;           // 64 floats = 32 float2 per row
    float2 hv = hp[(size_t)node * 32 + lane];
    float acc0 = bias[2 * lane]     + hv.x * wself;
    float acc1 = bias[2 * lane + 1] + hv.y * wself;
    int beg = rowptr[node], end = rowptr[node + 1];
    for (int e = beg; e < end; ++e) {
        int2  pe  = cedge[e];
        float wgt = __int_as_float(pe.y);
        float2 v  = hp[(size_t)pe.x * 32 + lane];
        acc0 = fmaf(v.x, wgt, acc0);
        acc1 = fmaf(v.y, wgt, acc1);
    }
    float2 o; o.x = acc0; o.y = acc1;
    ((float2*)OUT)[(size_t)node * 32 + lane] = o;
}

// H has row stride NC_PAD (48); only the first NCLASS (40) columns are real.
__global__ void k_agg40(const float* __restrict__ H, const int* __restrict__ rowptr,
                        const int2* __restrict__ cedge, const float* __restrict__ dinv,
                        const float* __restrict__ bias, float* __restrict__ OUT, int n) {
    int node = (blockIdx.x * blockDim.x + threadIdx.x) >> 5;
    int lane = threadIdx.x & 31;
    if (node >= n) return;
    node = __builtin_amdgcn_readfirstlane(node);
    float di = dinv[node];
    float wself = di * di;
    const bool hi = lane < 8;                      // 40 = 32 + 8
    const float* hrow = H + (size_t)node * NC_PAD;
    float acc0 = bias[lane] + hrow[lane] * wself;
    float acc1 = hi ? (bias[32 + lane] + hrow[32 + lane] * wself) : 0.f;
    int beg = rowptr[node], end = rowptr[node + 1];
    for (int e = beg; e < end; ++e) {
        int2  pe  = cedge[e];
        float wgt = __int_as_float(pe.y);
        const float* hs = H + (size_t)pe.x * NC_PAD;
        acc0 = fmaf(hs[lane], wgt, acc0);
        if (hi) acc1 = fmaf(hs[32 + lane], wgt, acc1);
    }
    float* orow = OUT + (size_t)node * NCLASS;
    orow[lane] = acc0;
    if (hi) orow[32 + lane] = acc1;
}

// ---------------------------------------------------------------- log_softmax
__global__ void k_logsoftmax(const float* __restrict__ in, float* __restrict__ out, int n) {
    int node = (blockIdx.x * blockDim.x + threadIdx.x) >> 5;
    int lane = threadIdx.x & 31;
    if (node >= n) return;
    const float* row = in + (size_t)node * NCLASS;
    float v0 = row[lane];
    float v1 = (lane < 8) ? row[32 + lane] : -INFINITY;
    float m = fmaxf(v0, v1);
    for (int o = 16; o > 0; o >>= 1) m = fmaxf(m, __shfl_xor(m, o, 32));
    float s = expf(v0 - m) + ((lane < 8) ? expf(v1 - m) : 0.f);
    for (int o = 16; o > 0; o >>= 1) s += __shfl_xor(s, o, 32);
    float ls = logf(s);
    float* orow = out + (size_t)node * NCLASS;
    orow[lane] = v0 - m - ls;
    if (lane < 8) orow[32 + lane] = v1 - m - ls;
}

// ---------------------------------------------------------------- launcher
extern "C" void kernel_launch(void* const* d_in, const int* in_sizes, int n_in,
                              void* d_out, int out_size, void* d_ws, size_t ws_size,
                              hipStream_t stream) {
    const float* x    = (const float*)d_in[0];
    const int*   eidx = (const int*)d_in[1];
    const int*   src  = eidx;
    const int*   dst  = eidx + N_EDGES;
    const float* W1 = (const float*)d_in[2];
    const float* b1 = (const float*)d_in[3];
    const float* W2 = (const float*)d_in[4];
    const float* b2 = (const float*)d_in[5];
    const float* W3 = (const float*)d_in[6];
    const float* b3 = (const float*)d_in[7];
    float* out = (float*)d_out;

    // carve workspace (aligned to 256B)
    char* w = (char*)d_ws;
    size_t off = 0;
    auto carve = [&](size_t bytes) -> void* {
        void* p = w + off;
        off = (off + bytes + 255) & ~(size_t)255;
        return p;
    };
    int*   cnt    = (int*)carve((size_t)N_NODES * 4);
    int*   rowptr = (int*)carve((size_t)(N_NODES + 1) * 4);
    int*   bsum   = (int*)carve(256 * 4);
    int*   cursor = (int*)carve((size_t)N_NODES * 4);
    float* dinv   = (float*)carve((size_t)N_NODES * 4);
    int2*  cedge  = (int2*)carve((size_t)N_EDGES * 8);
    float* Wp1    = (float*)carve((size_t)NFEAT * NHID * 4);
    float* Wp2    = (float*)carve((size_t)NHID * NHID * 4);
    float* Wp3    = (float*)carve((size_t)NHID * NC_PAD * 4);
    float* hbuf   = (float*)carve((size_t)N_NODES * NHID * 4);
    float* abuf   = (float*)carve((size_t)N_NODES * NHID * 4);

    const int nScanBlocks = (N_NODES + 1023) / 1024;   // 98

    // ---- graph structure: degrees -> rowptr -> CSR with norm weights
    k_zero_i32<<<(N_NODES + 255) / 256, 256, 0, stream>>>(cnt, N_NODES);
    k_hist<<<(N_EDGES + 255) / 256, 256, 0, stream>>>(dst, cnt, N_EDGES);
    k_scan_block<<<nScanBlocks, 1024, 0, stream>>>(cnt, rowptr, bsum, N_NODES);
    k_scan_sums<<<1, 32, 0, stream>>>(bsum, nScanBlocks);
    k_scan_add<<<(N_NODES + 1 + 255) / 256, 256, 0, stream>>>(rowptr, bsum, N_NODES, N_EDGES);
    k_prep<<<(N_NODES + 255) / 256, 256, 0, stream>>>(cnt, rowptr, dinv, cursor, N_NODES);
    k_fill<<<(N_EDGES + 255) / 256, 256, 0, stream>>>(src, dst, dinv, cursor, cedge, N_EDGES);

    // ---- pack weights into B-fragment order (layer 3 zero-padded to 48 cols)
    k_packW<<<(NFEAT * NHID + 255) / 256, 256, 0, stream>>>(W1, Wp1, NFEAT, NHID, NHID);
    k_packW<<<(NHID * NHID + 255) / 256, 256, 0, stream>>>(W2, Wp2, NHID, NHID, NHID);
    k_packW<<<(NHID * NC_PAD + 255) / 256, 256, 0, stream>>>(W3, Wp3, NHID, NCLASS, NC_PAD);

    constexpr int MGROUPS = N_NODES / 80;              // 1250 groups of 5x16 rows
    // ---- layer 1: h = x @ W1 ; agg + b1  (relu fused into next GEMM's A-load)
    {
        int waves = MGROUPS * (NHID / 16);             // 5000
        k_gemm_wmma<NFEAT, NHID, false><<<(waves + 7) / 8, 256, 0, stream>>>(x, Wp1, hbuf);
        k_agg64<<<N_NODES / 8, 256, 0, stream>>>(hbuf, rowptr, cedge, dinv, b1, abuf, N_NODES);
    }
    // ---- layer 2: h = relu(a1) @ W2 ; agg + b2
    {
        int waves = MGROUPS * (NHID / 16);
        k_gemm_wmma<NHID, NHID, true><<<(waves + 7) / 8, 256, 0, stream>>>(abuf, Wp2, hbuf);
        k_agg64<<<N_NODES / 8, 256, 0, stream>>>(hbuf, rowptr, cedge, dinv, b2, abuf, N_NODES);
    }
    // ---- layer 3: h = relu(a2) @ W3 (48 padded cols) ; agg + b3 ; log_softmax
    {
        int waves = MGROUPS * (NC_PAD / 16);           // 3750
        k_gemm_wmma<NHID, NC_PAD, true><<<(waves + 7) / 8, 256, 0, stream>>>(abuf, Wp3, hbuf);
        k_agg40<<<N_NODES / 8, 256, 0, stream>>>(hbuf, rowptr, cedge, dinv, b3, abuf, N_NODES);
        k_logsoftmax<<<N_NODES / 8, 256, 0, stream>>>(abuf, out, N_NODES);
    }
}